// BahdanauAttention_56195352101200
// MI455X (gfx1250) — compile-verified
//
#include <hip/hip_runtime.h>

typedef __attribute__((ext_vector_type(2))) float v2f;
typedef __attribute__((ext_vector_type(8))) float v8f;
typedef __attribute__((ext_vector_type(4))) unsigned int u32x4;
typedef __attribute__((ext_vector_type(8))) unsigned int u32x8;

#define B_DIM  32
#define TQ     4
#define S_DIM  2048
#define H_DIM  1024
#define K_DIM  2048        // 2H (keys feature dim, GEMM K)
#define BLK_M  64          // s-rows per workgroup in scores kernel
#define K_C    128         // K-chunk staged in LDS
#define NCHUNK (K_DIM / K_C)
#define LDS_STRIDE (K_C + 4)   // pad: +4 dwords -> bank-conflict-free ds_load_b64

// ---------------------------------------------------------------------------
// K1: wqb[b][h] = q[b,-1,:] . Wa_w[h,:] + Wa_b[h] + Ua_b[h]   (tiny GEMV)
// ---------------------------------------------------------------------------
__global__ __launch_bounds__(256) void wq_kernel(
    const float* __restrict__ q, const float* __restrict__ Wa_w,
    const float* __restrict__ Wa_b, const float* __restrict__ Ua_b,
    float* __restrict__ wqb) {
  const int b = blockIdx.x;
  __shared__ float qs[H_DIM];
  for (int i = threadIdx.x; i < H_DIM; i += 256)
    qs[i] = q[((size_t)b * TQ + (TQ - 1)) * H_DIM + i];
  __syncthreads();
  for (int h = threadIdx.x; h < H_DIM; h += 256) {
    const float* w = Wa_w + (size_t)h * H_DIM;
    float acc = 0.f;
#pragma unroll 4
    for (int d = 0; d < H_DIM; d += 4) {
      acc += qs[d] * w[d] + qs[d + 1] * w[d + 1] +
             qs[d + 2] * w[d + 2] + qs[d + 3] * w[d + 3];
    }
    wqb[b * H_DIM + h] = acc + Wa_b[h] + Ua_b[h];
  }
}

// ---------------------------------------------------------------------------
// TDM: issue a 2D tensor_load_to_lds of a BLK_M x K_C fp32 tile.
//   rows strided by B_DIM*K_DIM elements in global memory,
//   LDS rows padded 128 dwords -> +4 dwords (pad_interval=6, pad_amount=3)
//   so the landed layout has stride LDS_STRIDE = 132 floats.
// D# per CDNA5 ISA (group0: 4 SGPRs, group1: 8 SGPRs).
// ---------------------------------------------------------------------------
__device__ __forceinline__ void tdm_load_tile(const float* gbase,
                                              unsigned int lds_off) {
  const unsigned long long ga = (unsigned long long)(uintptr_t)gbase;
  u32x4 g0;
  g0[0] = 1u;                                    // count=1, user mode
  g0[1] = lds_off;                               // lds_addr (bytes)
  g0[2] = (unsigned int)ga;                      // global_addr[31:0]
  g0[3] = (unsigned int)(ga >> 32) | (2u << 30); // global_addr[56:32] | type=2

  u32x8 g1;
  g1[0] = (2u << 16)      // data_size = 4 bytes
        | (1u << 20)      // pad_enable
        | (6u << 22)      // pad_interval: 2^(6+1)=128 dwords
        | (3u << 25);     // pad_amount: 3+1=4 dwords
  g1[1] = ((unsigned int)K_DIM & 0xFFFFu) << 16;           // tensor_dim0 lo16
  g1[2] = ((unsigned int)K_DIM >> 16)                      // tensor_dim0 hi16
        | (((unsigned int)S_DIM & 0xFFFFu) << 16);         // tensor_dim1 lo16
  g1[3] = ((unsigned int)S_DIM >> 16)                      // tensor_dim1 hi16
        | ((unsigned int)K_C << 16);                       // tile_dim0 = 128
  g1[4] = (unsigned int)BLK_M;                             // tile_dim1=64, tile_dim2=0
  g1[5] = (unsigned int)(B_DIM * K_DIM);                   // tensor_dim0_stride lo32
  g1[6] = 0u;                                              // stride hi, dim1_stride lo
  g1[7] = 0u;

  // 2-operand form: VADDR2/VADDR3 = NULL (<=2D tensor)
  asm volatile("tensor_load_to_lds %0, %1" :: "s"(g0), "s"(g1) : "memory");
}

// ---------------------------------------------------------------------------
// K2: fused scores. score[b,s] = Va_b + sum_h Va_w[h]*tanh(wqb[b,h] + k[b,s,:].Ua_w[h,:])
// 512 threads = 16 waves, 4(M) x 4(N) wave grid.
//   - A (keys) tile TDM-loaded into double-buffered LDS; keys read ONCE from HBM
//   - full N=1024 in accumulators (16 wmma tiles/wave = 128 VGPRs)
//   - B (Ua_w, 8 MiB) streamed from L2 per-wave
// ---------------------------------------------------------------------------
__global__ __launch_bounds__(512) void scores_kernel(
    const float* __restrict__ keys, const float* __restrict__ Ua,
    const float* __restrict__ wqb, const float* __restrict__ Va_w,
    const float* __restrict__ Va_b, float* __restrict__ scores) {
  __shared__ float Atile[2][BLK_M * LDS_STRIDE];
  __shared__ float score_lds[BLK_M];

  const int b    = blockIdx.x;        // batch
  const int s0   = blockIdx.y * BLK_M;
  const int tid  = threadIdx.x;
  const int lane = tid & 31;
  const int wave = tid >> 5;
  const int wm   = wave & 3;          // m-wave: rows [wm*16, wm*16+16)
  const int wn   = wave >> 2;         // n-wave: cols [wn*256, wn*256+256)
  const int l15  = lane & 15;
  const int lhi  = lane >> 4;         // 0|1 -> K sub-pair / row half

  v8f acc[16];
#pragma unroll
  for (int t = 0; t < 16; ++t)
#pragma unroll
    for (int j = 0; j < 8; ++j) acc[t][j] = 0.f;

  if (tid < BLK_M) score_lds[tid] = 0.f;

  const unsigned int lds0 = (unsigned int)(uintptr_t)(void*)&Atile[0][0];
  const unsigned int lds1 = (unsigned int)(uintptr_t)(void*)&Atile[1][0];
  const float* gtile = keys + ((size_t)s0 * B_DIM + b) * K_DIM;

  if (wave == 0) tdm_load_tile(gtile, lds0);          // prologue: chunk 0

  for (int c = 0; c < NCHUNK; ++c) {
    if (wave == 0) {
      if (c + 1 < NCHUNK) {
        // prefetch next chunk into the other buffer, keep it in flight
        tdm_load_tile(gtile + (c + 1) * K_C, (c & 1) ? lds0 : lds1);
        __builtin_amdgcn_s_wait_tensorcnt(1);   // chunk c landed
      } else {
        __builtin_amdgcn_s_wait_tensorcnt(0);
      }
    }
    __syncthreads();   // LDS chunk c visible to all waves

    // A fragment (16x4 f32): lanes0-15 M=l15 K={kk,kk+1}; lanes16-31 K={kk+2,kk+3}
    const float* Arow = &Atile[c & 1][(wm * 16 + l15) * LDS_STRIDE + 2 * lhi];
    // B fragment (4x16 f32): B[k][n] = Ua_w[n][k]; lane's K-pair contiguous in Ua row
    const float* Bbase = Ua + ((size_t)(wn * 256 + l15)) * K_DIM + c * K_C + 2 * lhi;

    for (int kk = 0; kk < K_C; kk += 4) {
      v2f a = *(const v2f*)(Arow + kk);
#pragma unroll
      for (int t = 0; t < 16; ++t) {
        v2f bf = *(const v2f*)(Bbase + (size_t)t * 16 * K_DIM + kk);
        acc[t] = __builtin_amdgcn_wmma_f32_16x16x4_f32(
            false, a, false, bf, (short)0, acc[t], false, false);
      }
    }
    __syncthreads();   // all reads of this buffer done before TDM reuses it
  }

  // Epilogue: tanh + Va_w dot, reduce over this wave's 256 columns
  float sp[8];
#pragma unroll
  for (int j = 0; j < 8; ++j) sp[j] = 0.f;
#pragma unroll
  for (int t = 0; t < 16; ++t) {
    const int n     = wn * 256 + t * 16 + l15;   // this lane's column
    const float vw  = Va_w[n];
    const float wqv = wqb[b * H_DIM + n];
#pragma unroll
    for (int j = 0; j < 8; ++j) {                // VGPR j: row = lhi*8 + j
      sp[j] += vw * tanhf(acc[t][j] + wqv);
    }
  }
#pragma unroll
  for (int j = 0; j < 8; ++j)
    atomicAdd(&score_lds[wm * 16 + lhi * 8 + j], sp[j]);   // ds_add_f32
  __syncthreads();
  if (tid < BLK_M)
    scores[b * S_DIM + s0 + tid] = score_lds[tid] + Va_b[0];
}

// ---------------------------------------------------------------------------
// K3: per-batch softmax over S=2048
// ---------------------------------------------------------------------------
__global__ __launch_bounds__(256) void softmax_kernel(
    const float* __restrict__ scores, float* __restrict__ weights) {
  const int b = blockIdx.x;
  __shared__ float red[256];
  __shared__ float stats[2];
  const float* src = scores + b * S_DIM;

  float m = -3.4e38f;
  for (int i = threadIdx.x; i < S_DIM; i += 256) m = fmaxf(m, src[i]);
  red[threadIdx.x] = m;
  __syncthreads();
  for (int off = 128; off > 0; off >>= 1) {
    if (threadIdx.x < off)
      red[threadIdx.x] = fmaxf(red[threadIdx.x], red[threadIdx.x + off]);
    __syncthreads();
  }
  if (threadIdx.x == 0) stats[0] = red[0];
  __syncthreads();
  m = stats[0];

  float s = 0.f;
  for (int i = threadIdx.x; i < S_DIM; i += 256) s += __expf(src[i] - m);
  red[threadIdx.x] = s;
  __syncthreads();
  for (int off = 128; off > 0; off >>= 1) {
    if (threadIdx.x < off)
      red[threadIdx.x] += red[threadIdx.x + off];
    __syncthreads();
  }
  if (threadIdx.x == 0) stats[1] = 1.f / red[0];
  __syncthreads();
  const float inv = stats[1];
  for (int i = threadIdx.x; i < S_DIM; i += 256)
    weights[b * S_DIM + i] = __expf(src[i] - m) * inv;
}

// ---------------------------------------------------------------------------
// K4: context[b,d] = sum_s weights[b,s] * keys[s,b,d]   (second keys pass)
// ---------------------------------------------------------------------------
__global__ __launch_bounds__(256) void context_kernel(
    const float* __restrict__ keys, const float* __restrict__ weights,
    float* __restrict__ ctx) {
  const int b = blockIdx.x;
  const int d = blockIdx.y * 256 + threadIdx.x;
  __shared__ float wsh[S_DIM];
  for (int i = threadIdx.x; i < S_DIM; i += 256) wsh[i] = weights[b * S_DIM + i];
  __syncthreads();
  float a0 = 0.f, a1 = 0.f, a2 = 0.f, a3 = 0.f;
  for (int s = 0; s < S_DIM; s += 4) {
    a0 += wsh[s]     * keys[((size_t)(s)     * B_DIM + b) * K_DIM + d];
    a1 += wsh[s + 1] * keys[((size_t)(s + 1) * B_DIM + b) * K_DIM + d];
    a2 += wsh[s + 2] * keys[((size_t)(s + 2) * B_DIM + b) * K_DIM + d];
    a3 += wsh[s + 3] * keys[((size_t)(s + 3) * B_DIM + b) * K_DIM + d];
  }
  ctx[b * K_DIM + d] = (a0 + a1) + (a2 + a3);
}

// ---------------------------------------------------------------------------
extern "C" void kernel_launch(void* const* d_in, const int* in_sizes, int n_in,
                              void* d_out, int out_size, void* d_ws, size_t ws_size,
                              hipStream_t stream) {
  const float* query = (const float*)d_in[0];
  const float* keys  = (const float*)d_in[1];
  const float* Wa_w  = (const float*)d_in[2];
  const float* Wa_b  = (const float*)d_in[3];
  const float* Ua_w  = (const float*)d_in[4];
  const float* Ua_b  = (const float*)d_in[5];
  const float* Va_w  = (const float*)d_in[6];
  const float* Va_b  = (const float*)d_in[7];

  float* out     = (float*)d_out;
  float* ctx     = out;                        // (32, 1, 2048)
  float* weights = out + B_DIM * K_DIM;        // (32, 1, 2048)

  float* wqb    = (float*)d_ws;                // 32*1024 floats
  float* scores = wqb + B_DIM * H_DIM;         // 32*2048 floats

  wq_kernel<<<B_DIM, 256, 0, stream>>>(query, Wa_w, Wa_b, Ua_b, wqb);
  scores_kernel<<<dim3(B_DIM, S_DIM / BLK_M), 512, 0, stream>>>(
      keys, Ua_w, wqb, Va_w, Va_b, scores);
  softmax_kernel<<<B_DIM, 256, 0, stream>>>(scores, weights);
  context_kernel<<<dim3(B_DIM, K_DIM / 256), 256, 0, stream>>>(keys, weights, ctx);
}